// PyramidalAttention_37022618091570
// MI455X (gfx1250) — compile-verified
//
#include <hip/hip_runtime.h>
#include <hip/hip_bf16.h>

// Pyramidal attention for MI455X (gfx1250, wave32).
//  k1: QKV projection GEMMs via V_WMMA_F32_16X16X4_F32 (K=7 padded to 8,
//      pad handled with clamped-address loads + mask, no EXEC branches)
//  k2: gathered attention (L2-resident K/V gathers + global_prefetch on V,
//      wave32 shuffle softmax)
//  k3: output projection GEMM (K=512, 128 chained WMMA, b64 fragment loads)
//      + residual + LayerNorm

#define B_ 2
#define L_ 4096
#define C_ 7
#define H_ 8
#define D_ 64
#define W_ 512          // H_*D_
#define M_ 32
#define ROWS_ (B_*L_)   // 8192

typedef float v2f __attribute__((ext_vector_type(2)));
typedef float v8f __attribute__((ext_vector_type(8)));

// ---------------------------------------------------------------------------
// Kernel 1: q/k/v = hs @ w{q,k,v}^T  (q additionally scaled by 1/sqrt(64))
// One wave per 16x16 output tile. A = hs tile (16x4), B = w^T tile (4x16).
// K = 7 padded to 8 -> two WMMA steps. The K=7 pad element is produced by
// loading a clamped in-range address and multiplying by a 0/1 mask, so the
// fragment loads stay unconditional (no saveexec around the WMMA inputs).
// ---------------------------------------------------------------------------
__global__ __launch_bounds__(256) void qkv_proj_wmma(
    const float* __restrict__ hs,
    const float* __restrict__ wq,
    const float* __restrict__ wk,
    const float* __restrict__ wv,
    float* __restrict__ q, float* __restrict__ k, float* __restrict__ v)
{
    const int MT = ROWS_ / 16;   // 512 m-tiles
    const int NT = W_ / 16;      // 32 n-tiles
    const int lane = threadIdx.x & 31;
    const int wave = threadIdx.x >> 5;
    const int tile = blockIdx.x * (blockDim.x >> 5) + wave; // [0, 3*MT*NT)

    const int mat = tile / (MT * NT);        // 0=q 1=k 2=v (wave-uniform)
    const int t   = tile % (MT * NT);
    const int mt  = t / NT;
    const int nt  = t % NT;

    const float* __restrict__ wmat = (mat == 0) ? wq : (mat == 1) ? wk : wv;
    float* __restrict__ outp       = (mat == 0) ? q  : (mat == 1) ? k  : v;

    // A-matrix 16x4 fp32 layout: lanes 0-15 -> K={0,1}, lanes 16-31 -> K={2,3}
    const int mrow  = lane & 15;
    const int khalf = lane >> 4;
    const int arow  = mt * 16 + mrow;          // global row (b*L + l)
    const int ncol  = nt * 16 + (lane & 15);   // global output column

    v8f c = {};
#pragma unroll
    for (int ks = 0; ks < 2; ++ks) {
        const int   k0   = ks * 4 + khalf * 2;            // 0,2,4,6
        const bool  inr  = (k0 + 1 < C_);                 // false only at K=7
        const int   k1i  = inr ? (k0 + 1) : 0;            // clamped (in-range)
        const float kmsk = inr ? 1.0f : 0.0f;
        v2f a, b;
        a.x = hs[arow * C_ + k0];
        a.y = hs[arow * C_ + k1i] * kmsk;
        b.x = wmat[ncol * C_ + k0];
        b.y = wmat[ncol * C_ + k1i] * kmsk;
        c = __builtin_amdgcn_wmma_f32_16x16x4_f32(
                false, a, false, b, (short)0, c, false, false);
    }

    const float scale = (mat == 0) ? 0.125f : 1.0f;   // 1/sqrt(D_)
    // D layout: VGPR r -> M = r (lanes 0-15) / r+8 (lanes 16-31), N = lane&15
    const int rbase = mt * 16 + (khalf ? 8 : 0);
#pragma unroll
    for (int r = 0; r < 8; ++r)
        outp[(rbase + r) * W_ + ncol] = c[r] * scale;
}

// ---------------------------------------------------------------------------
// Kernel 2: gathered attention. Block = one (b,l), wave h = head h.
// Scores: lane m owns gathered key m (dot over D=64, q from LDS).
// Softmax: wave32 __shfl_xor reductions. Out: lane owns channels d, d+32.
// V rows are prefetched (global_prefetch_b8) as soon as indices are known,
// so the L2 gather latency overlaps the score/softmax phase.
// ---------------------------------------------------------------------------
__global__ __launch_bounds__(256) void pyr_attn(
    const float* __restrict__ q, const float* __restrict__ k,
    const float* __restrict__ v, const int* __restrict__ qk_mask,
    float* __restrict__ aout)
{
    const int bl = blockIdx.x;          // 0..ROWS_-1
    const int b  = bl / L_;
    const int l  = bl % L_;

    __shared__ float qs[W_];
    __shared__ int   idxs[M_];

    const int tid = threadIdx.x;
    qs[tid]       = q[bl * W_ + tid];
    qs[tid + 256] = q[bl * W_ + tid + 256];
    if (tid < M_) idxs[tid] = qk_mask[l * M_ + tid];
    __syncthreads();

    const int h    = tid >> 5;          // head
    const int lane = tid & 31;          // key index m

    const int  raw   = idxs[lane];
    const bool valid = (raw >= 0);
    const int  krow  = b * L_ + (valid ? raw : 0);

    // Prefetch this lane's gathered V row (256 B span -> 2 cachelines) so it
    // is L0/WGP$-resident by the time the weighted-V accumulation runs.
    const float* __restrict__ vpre = v + (size_t)krow * W_ + h * D_;
    __builtin_prefetch(vpre,      0, 0);
    __builtin_prefetch(vpre + 32, 0, 0);

    // ---- scores (q pre-scaled by 1/sqrt(D)) ----
    const float* __restrict__ kp = k + (size_t)krow * W_ + h * D_;
    float s = 0.0f;
#pragma unroll 8
    for (int d = 0; d < D_; ++d)
        s = fmaf(qs[h * D_ + d], kp[d], s);
    float score = valid ? s : -1.0e9f;

    // ---- softmax over 32 keys (wave32 reductions) ----
    float mx = score;
#pragma unroll
    for (int off = 16; off > 0; off >>= 1)
        mx = fmaxf(mx, __shfl_xor(mx, off, 32));
    float p = valid ? __expf(score - mx) : 0.0f;
    float sum = p;
#pragma unroll
    for (int off = 16; off > 0; off >>= 1)
        sum += __shfl_xor(sum, off, 32);
    const float attn = p * (1.0f / fmaxf(sum, 1.0e-30f));

    // ---- out[h,d] = sum_m attn[m] * v[idx[m], h, d]; lane owns d, d+32 ----
    float a0 = 0.0f, a1 = 0.0f;
#pragma unroll 4
    for (int m = 0; m < M_; ++m) {
        const float am  = __shfl(attn, m, 32);
        const int  raw2 = idxs[m];
        const int  vrow = b * L_ + (raw2 >= 0 ? raw2 : 0);
        const float* __restrict__ vp = v + (size_t)vrow * W_ + h * D_;
        a0 = fmaf(am, vp[lane],      a0);
        a1 = fmaf(am, vp[lane + 32], a1);
    }
    aout[bl * W_ + h * D_ + lane]      = a0;
    aout[bl * W_ + h * D_ + lane + 32] = a1;
}

// ---------------------------------------------------------------------------
// Kernel 3: context = aout @ fc_w^T + fc_b + residual, then LayerNorm(C=7).
// One wave per 16-row tile; K=512 -> 128 chained V_WMMA_F32_16X16X4_F32.
// Fragments are loaded as 8-byte v2f (k0 even, 2048 B row strides -> aligned)
// so each WMMA step costs two global_load_b64. Columns 7..15 use a 0/1 mask
// on the B fragment (no EXEC divergence before the WMMA chain).
// ---------------------------------------------------------------------------
__global__ __launch_bounds__(256) void out_proj_ln_wmma(
    const float* __restrict__ aout, const float* __restrict__ fc_w,
    const float* __restrict__ fc_b, const float* __restrict__ hs,
    const float* __restrict__ ln_s, const float* __restrict__ ln_b,
    float* __restrict__ out)
{
    const int lane = threadIdx.x & 31;
    const int wave = threadIdx.x >> 5;
    const int mt   = blockIdx.x * (blockDim.x >> 5) + wave;  // 0..511

    const int mrow  = lane & 15;
    const int khalf = lane >> 4;
    const int arow  = mt * 16 + mrow;
    const int col   = lane & 15;              // fc output channel (valid < 7)
    const bool cv   = (col < C_);

    const float cmask = cv ? 1.0f : 0.0f;
    const int   bcol  = cv ? col : 0;         // clamped row of fc_w
    const float* __restrict__ arowp = aout + (size_t)arow * W_;
    const float* __restrict__ browp = fc_w + (size_t)bcol * W_;

    v8f c = {};
    for (int kk = 0; kk < W_; kk += 4) {
        const int k0 = kk + khalf * 2;        // even -> 8B-aligned v2f loads
        v2f a = *(const v2f*)(arowp + k0);
        v2f b = *(const v2f*)(browp + k0);
        b.x *= cmask;
        b.y *= cmask;
        c = __builtin_amdgcn_wmma_f32_16x16x4_f32(
                false, a, false, b, (short)0, c, false, false);
    }

    const float bias = cv ? fc_b[col] : 0.0f;
    const float lsc  = cv ? ln_s[col] : 0.0f;
    const float lbi  = cv ? ln_b[col] : 0.0f;
    const int   rbase = mt * 16 + (khalf ? 8 : 0);

#pragma unroll
    for (int r = 0; r < 8; ++r) {
        const int   row = rbase + r;
        const float val = c[r] + bias + (cv ? hs[row * C_ + col] : 0.0f);

        // LayerNorm over the 7 valid columns. Shuffle masks {8,4,2,1} keep
        // the two 16-lane halves (rows rbase.. vs rbase+8..) independent.
        float x = cv ? val : 0.0f;
        float ssum = x;
#pragma unroll
        for (int off = 8; off > 0; off >>= 1)
            ssum += __shfl_xor(ssum, off, 32);
        const float mu = ssum * (1.0f / (float)C_);

        float dv = cv ? (val - mu) : 0.0f;
        float vsum = dv * dv;
#pragma unroll
        for (int off = 8; off > 0; off >>= 1)
            vsum += __shfl_xor(vsum, off, 32);
        const float var = vsum * (1.0f / (float)C_);

        const float y = (val - mu) * rsqrtf(var + 1.0e-6f) * lsc + lbi;
        if (cv) out[row * C_ + col] = y;
    }
}

// ---------------------------------------------------------------------------
extern "C" void kernel_launch(void* const* d_in, const int* in_sizes, int n_in,
                              void* d_out, int out_size, void* d_ws, size_t ws_size,
                              hipStream_t stream)
{
    const float* hs  = (const float*)d_in[0];
    const float* wq  = (const float*)d_in[1];
    const float* wk  = (const float*)d_in[2];
    const float* wv  = (const float*)d_in[3];
    const float* fcw = (const float*)d_in[4];
    const float* fcb = (const float*)d_in[5];
    const float* lns = (const float*)d_in[6];
    const float* lnb = (const float*)d_in[7];
    const int*   qkm = (const int*)d_in[8];
    // d_in[9] (k_q_mask) is unused by the reference.
    float* out = (float*)d_out;

    // Workspace: q, k, v, attn_out — 4 x ROWS_*W_ floats = 64 MB total.
    float* qb = (float*)d_ws;
    float* kb = qb + (size_t)ROWS_ * W_;
    float* vb = kb + (size_t)ROWS_ * W_;
    float* ab = vb + (size_t)ROWS_ * W_;

    // k1: 3 * 512 * 32 = 49152 tiles, 8 waves/block -> 6144 blocks (exact).
    qkv_proj_wmma<<<6144, 256, 0, stream>>>(hs, wq, wk, wv, qb, kb, vb);
    // k2: one block per (b,l).
    pyr_attn<<<ROWS_, 256, 0, stream>>>(qb, kb, vb, qkm, ab);
    // k3: 512 m-tiles, 8 waves/block -> 64 blocks (exact).
    out_proj_ln_wmma<<<64, 256, 0, stream>>>(ab, fcw, fcb, hs, lns, lnb, out);
}